// GraphNet_19877108646002
// MI455X (gfx1250) — compile-verified
//
#include <hip/hip_runtime.h>
#include <hip/hip_bf16.h>
#include <math.h>

typedef __attribute__((ext_vector_type(2))) float v2f;
typedef __attribute__((ext_vector_type(4))) float v4f;
typedef __attribute__((ext_vector_type(8))) float v8f;

#define USE_ASYNC_STAGE 1

#if USE_ASYNC_STAGE
// gfx1250 async global->LDS copy (ASYNCcnt-tracked), CDNA5 ISA
__device__ __forceinline__ void async_copy_b128(const float* gp, float* lp) {
  unsigned lds = (unsigned)(unsigned long long)(__attribute__((address_space(3))) float*)lp;
  unsigned long long ga = (unsigned long long)gp;
  asm volatile("global_load_async_to_lds_b128 %0, %1, off" :: "v"(lds), "v"(ga) : "memory");
}
__device__ __forceinline__ void async_wait_all() {
  asm volatile("s_wait_asynccnt 0" ::: "memory");
}
#endif

// ---------------------------------------------------------------------------
// CSR construction: count -> scan -> fill
// ---------------------------------------------------------------------------
__global__ __launch_bounds__(256) void gnet_count(const int* __restrict__ col,
                                                  int* __restrict__ cnt, int E) {
  int e = blockIdx.x * 256 + threadIdx.x;
  if (e < E) atomicAdd(&cnt[col[e]], 1);
}

__global__ __launch_bounds__(256) void gnet_scan(const int* __restrict__ cnt,
                                                 int* __restrict__ rowptr,
                                                 int* __restrict__ cursor,
                                                 float* __restrict__ deginv, int n) {
  __shared__ int sbuf[256];
  __shared__ int srun;
  const int tid = threadIdx.x;
  if (tid == 0) srun = 0;
  __syncthreads();
  for (int base = 0; base < n; base += 256) {
    int i = base + tid;
    int v = (i < n) ? cnt[i] : 0;
    sbuf[tid] = v;
    __syncthreads();
    #pragma unroll
    for (int off = 1; off < 256; off <<= 1) {
      int t = (tid >= off) ? sbuf[tid - off] : 0;
      __syncthreads();
      sbuf[tid] += t;
      __syncthreads();
    }
    const int incl = sbuf[tid];
    const int excl = incl - v;
    const int run  = srun;
    if (i < n) {
      rowptr[i] = run + excl;
      cursor[i] = run + excl;
      deginv[i] = (v > 0) ? (1.0f / (float)v) : 0.0f;
    }
    __syncthreads();
    if (tid == 255) srun = run + incl;
    __syncthreads();
  }
  if (tid == 0) rowptr[n] = srun;
}

__global__ __launch_bounds__(256) void gnet_fill(const int* __restrict__ row,
                                                 const int* __restrict__ col,
                                                 int* __restrict__ cursor,
                                                 int* __restrict__ csr_src,
                                                 int* __restrict__ csr_eid, int E) {
  int e = blockIdx.x * 256 + threadIdx.x;
  if (e < E) {
    int slot = atomicAdd(&cursor[col[e]], 1);
    csr_src[slot] = row[e];
    csr_eid[slot] = e;
  }
}

// ---------------------------------------------------------------------------
// Gather-based aggregation (sum or mean) over incoming edges, vectorized:
// each thread owns 4 consecutive channels (float4 loads -> a wave pulls a
// full 512B source row per edge). Block = 128 threads; a node maps to one
// sub-group of H/4 threads (== 1 wave for H=128, 2 waves for H=256), so the
// edge-feature broadcast (emode, H=128 only) stays wave-local for __shfl.
// 2-edge unroll for memory-level parallelism; X is L2-resident (<=102MB<192MB).
// ---------------------------------------------------------------------------
template <int H>
__global__ __launch_bounds__(128) void gnet_agg(const float* __restrict__ X,
                                                const int* __restrict__ rowptr,
                                                const int* __restrict__ csr_src,
                                                const int* __restrict__ csr_eid,
                                                const float* __restrict__ edge_attr,
                                                const float* __restrict__ we1,
                                                const float* __restrict__ deginv,
                                                const float* __restrict__ bias,
                                                float* __restrict__ outA,  // pre-act (or null)
                                                float* __restrict__ outB,  // relu (or null)
                                                int Nn) {
  constexpr int TPN = H / 4;    // threads per node
  constexpr int NPB = 128 / TPN;  // nodes per block
  __shared__ v4f we1s[8 * 32];    // we1 as 8 x (128/4) float4 (emode only)
  const int tid  = threadIdx.x;
  const int sub  = tid / TPN;       // node slot in block
  const int t    = tid - sub * TPN; // channel group within node
  const int c4   = t * 4;
  const int lane = tid & 31;
  const bool emode = (csr_eid != nullptr);
  if (emode) {
    for (int i = tid; i < 8 * 32; i += 128) we1s[i] = ((const v4f*)we1)[i];
    __syncthreads();
  }
  const int v = blockIdx.x * NPB + sub;
  if (v >= Nn) return;

  const int beg = rowptr[v], end = rowptr[v + 1];
  v4f acc = {};
  int s = beg;
  for (; s + 1 < end; s += 2) {
    const int src0 = csr_src[s];
    const int src1 = csr_src[s + 1];
    v4f m0 = *(const v4f*)(X + (size_t)src0 * H + c4);
    v4f m1 = *(const v4f*)(X + (size_t)src1 * H + c4);
    if (emode) {
      const int e0 = csr_eid[s], e1 = csr_eid[s + 1];
      float ev0 = 0.0f, ev1 = 0.0f;
      if (lane < 8) {
        ev0 = edge_attr[(size_t)e0 * 8 + lane];
        ev1 = edge_attr[(size_t)e1 * 8 + lane];
      }
      #pragma unroll
      for (int k = 0; k < 8; ++k) {
        const float a0 = __shfl(ev0, k, 32);
        const float a1 = __shfl(ev1, k, 32);
        const v4f w = we1s[k * 32 + t];
        m0 += a0 * w;
        m1 += a1 * w;
      }
    }
    acc += m0 + m1;
  }
  if (s < end) {
    const int src0 = csr_src[s];
    v4f m0 = *(const v4f*)(X + (size_t)src0 * H + c4);
    if (emode) {
      const int e0 = csr_eid[s];
      float ev0 = 0.0f;
      if (lane < 8) ev0 = edge_attr[(size_t)e0 * 8 + lane];
      #pragma unroll
      for (int k = 0; k < 8; ++k) {
        m0 += __shfl(ev0, k, 32) * we1s[k * 32 + t];
      }
    }
    acc += m0;
  }
  if (deginv) acc *= deginv[v];
  if (bias)   acc += *(const v4f*)(bias + c4);
  float* pa = outA ? (outA + (size_t)v * H + c4) : nullptr;
  float* pb = outB ? (outB + (size_t)v * H + c4) : nullptr;
  if (pa) *(v4f*)pa = acc;
  if (pb) {
    v4f r;
    r.x = fmaxf(acc.x, 0.0f); r.y = fmaxf(acc.y, 0.0f);
    r.z = fmaxf(acc.z, 0.0f); r.w = fmaxf(acc.w, 0.0f);
    *(v4f*)pb = r;
  }
}

// ---------------------------------------------------------------------------
// Fused WMMA fp32 GEMM:  Y = act(X1@W1 [+ X2@W2] + bias) [+ Z]
// Block = 256 threads (8 waves), owns 16 rows x all M cols (grid.y==1 ->
// in-place X==Y safe: X tiles fully staged in LDS before any store).
// Staging via gfx1250 async global->LDS b128 (ASYNCcnt) when enabled.
// Inner loop: v_wmma_f32_16x16x4_f32 (fp32 in/out, matches reference math).
// ---------------------------------------------------------------------------
template <int K, int M>
__global__ __launch_bounds__(256) void gnet_gemm(const float* __restrict__ X1,
                                                 const float* __restrict__ Wt1,
                                                 const float* __restrict__ X2,
                                                 const float* __restrict__ Wt2,
                                                 const float* __restrict__ bias,
                                                 const float* __restrict__ Zres,
                                                 float* __restrict__ Y,
                                                 int Nn, int act) {
  constexpr int NCT    = (M + 127) / 128;  // col tiles per wave
  constexpr int LDSROW = K + 4;            // pad (mult of 4 -> b128-aligned, banks spread)
  constexpr int NV     = 16 * (K / 4);     // float4 elements per tile
  extern __shared__ float smem[];
  float* Xs1 = smem;
  float* Xs2 = smem + 16 * LDSROW;

  const int tid  = threadIdx.x;
  const int row0 = blockIdx.x * 16;
  const bool hasX2 = (X2 != nullptr);

#if USE_ASYNC_STAGE
  for (int i = tid; i < NV; i += 256) {
    const int r  = i / (K / 4);
    const int c4 = (i - r * (K / 4)) * 4;
    const int rr = row0 + r;
    if (rr < Nn) async_copy_b128(X1 + (size_t)rr * K + c4, Xs1 + r * LDSROW + c4);
  }
  if (hasX2) {
    for (int i = tid; i < NV; i += 256) {
      const int r  = i / (K / 4);
      const int c4 = (i - r * (K / 4)) * 4;
      const int rr = row0 + r;
      if (rr < Nn) async_copy_b128(X2 + (size_t)rr * K + c4, Xs2 + r * LDSROW + c4);
    }
  }
  async_wait_all();
#else
  for (int i = tid; i < NV; i += 256) {
    const int r  = i / (K / 4);
    const int c4 = (i - r * (K / 4)) * 4;
    const int rr = row0 + r;
    v4f val = {};
    if (rr < Nn) val = *(const v4f*)(X1 + (size_t)rr * K + c4);
    *(v4f*)(Xs1 + r * LDSROW + c4) = val;
  }
  if (hasX2) {
    for (int i = tid; i < NV; i += 256) {
      const int r  = i / (K / 4);
      const int c4 = (i - r * (K / 4)) * 4;
      const int rr = row0 + r;
      v4f val = {};
      if (rr < Nn) val = *(const v4f*)(X2 + (size_t)rr * K + c4);
      *(v4f*)(Xs2 + r * LDSROW + c4) = val;
    }
  }
#endif
  __syncthreads();

  const int wave = tid >> 5;
  const int lane = tid & 31;
  const int nn   = lane & 15;
  const int kb   = (lane >> 4) << 1;  // 0 or 2 (ISA 7.12.2, 32-bit A 16x4)
  const int mrow = lane & 15;

  if (wave * 16 >= M) return;  // wave-uniform: EXEC all-ones for WMMA

  v8f acc[NCT];
  int  cols[NCT];   // clamped, always in-bounds
  bool colok[NCT];
  #pragma unroll
  for (int j = 0; j < NCT; ++j) {
    acc[j] = {};
    int c  = wave * 16 + j * 128 + nn;
    colok[j] = (c < M);
    cols[j]  = colok[j] ? c : 0;  // junk cols never stored
  }

  for (int k = 0; k < K; k += 4) {
    v2f a;
    a.x = Xs1[mrow * LDSROW + k + kb];
    a.y = Xs1[mrow * LDSROW + k + kb + 1];
    #pragma unroll
    for (int j = 0; j < NCT; ++j) {
      v2f b;
      b.x = Wt1[(size_t)(k + kb) * M + cols[j]];
      b.y = Wt1[(size_t)(k + kb + 1) * M + cols[j]];
      acc[j] = __builtin_amdgcn_wmma_f32_16x16x4_f32(false, a, false, b,
                                                     (short)0, acc[j], false, false);
    }
  }
  if (hasX2) {
    for (int k = 0; k < K; k += 4) {
      v2f a;
      a.x = Xs2[mrow * LDSROW + k + kb];
      a.y = Xs2[mrow * LDSROW + k + kb + 1];
      #pragma unroll
      for (int j = 0; j < NCT; ++j) {
        v2f b;
        b.x = Wt2[(size_t)(k + kb) * M + cols[j]];
        b.y = Wt2[(size_t)(k + kb + 1) * M + cols[j]];
        acc[j] = __builtin_amdgcn_wmma_f32_16x16x4_f32(false, a, false, b,
                                                       (short)0, acc[j], false, false);
      }
    }
  }

  // Epilogue. C/D layout: VGPR i -> row (i + (lane>=16 ? 8 : 0)), col = lane&15.
  #pragma unroll
  for (int j = 0; j < NCT; ++j) {
    if (!colok[j]) continue;
    const float bv = bias ? bias[cols[j]] : 0.0f;
    #pragma unroll
    for (int i = 0; i < 8; ++i) {
      const int r = row0 + i + ((lane >= 16) ? 8 : 0);
      if (r >= Nn) continue;
      float v = acc[j][i] + bv;
      if (act == 1)      v = fmaxf(v, 0.0f);
      else if (act == 2) v = 1.0f / (1.0f + __expf(-v));
      if (Zres) v += Zres[(size_t)r * M + cols[j]];
      Y[(size_t)r * M + cols[j]] = v;
    }
  }
}

template <int K, int M>
static void launch_gemm(const float* X1, const float* Wt1,
                        const float* X2, const float* Wt2,
                        const float* bias, const float* Zres,
                        float* Y, int Nn, int act, hipStream_t s) {
  dim3 grid((Nn + 15) / 16, 1);
  size_t lds = (size_t)16 * (K + 4) * sizeof(float) * (X2 ? 2 : 1);
  gnet_gemm<K, M><<<grid, 256, lds, s>>>(X1, Wt1, X2, Wt2, bias, Zres, Y, Nn, act);
}

// ---------------------------------------------------------------------------
extern "C" void kernel_launch(void* const* d_in, const int* in_sizes, int n_in,
                              void* d_out, int out_size, void* d_ws, size_t ws_size,
                              hipStream_t stream) {
  const float* node_attr = (const float*)d_in[0];
  const float* edge_attr = (const float*)d_in[1];
  const int*   edge_idx  = (const int*)d_in[2];
  const float* wn1 = (const float*)d_in[3];
  const float* we1 = (const float*)d_in[4];
  const float* b1  = (const float*)d_in[5];
  const float* wn2 = (const float*)d_in[6];
  const float* b2  = (const float*)d_in[7];
  const float* sl0 = (const float*)d_in[8];
  const float* sr0 = (const float*)d_in[9];
  const float* sb0 = (const float*)d_in[10];
  const float* sl1 = (const float*)d_in[11];
  const float* sr1 = (const float*)d_in[12];
  const float* sb1 = (const float*)d_in[13];
  const float* sl2 = (const float*)d_in[14];
  const float* sr2 = (const float*)d_in[15];
  const float* sb2 = (const float*)d_in[16];
  const float* w3a = (const float*)d_in[17];
  const float* b3a = (const float*)d_in[18];
  const float* w3b = (const float*)d_in[19];
  const float* b3b = (const float*)d_in[20];

  const int N_ = in_sizes[0] / 16;  // 100000
  const int E_ = in_sizes[1] / 8;   // 1600000
  const int* rowp = edge_idx;        // sources
  const int* colp = edge_idx + E_;   // targets

  // Workspace carve-out
  char* ws = (char*)d_ws;
  size_t p = 0;
  auto carve = [&](size_t bytes) { void* q = ws + p; p += (bytes + 255) & ~(size_t)255; return q; };
  float* W0 = (float*)carve((size_t)N_ * 128 * 4);  // xn1 / y / m0 / x_final
  float* W1 = (float*)carve((size_t)N_ * 128 * 4);  // x / h
  float* W2 = (float*)carve((size_t)N_ * 128 * 4);  // x_temp
  float* W3 = (float*)carve((size_t)N_ * 256 * 4);  // x1 / m3
  float* W4 = (float*)carve((size_t)N_ * 256 * 4);  // m2 / x2
  int*   cnt     = (int*)carve((size_t)N_ * 4);
  int*   rowptr  = (int*)carve(((size_t)N_ + 1) * 4);
  int*   cursor  = (int*)carve((size_t)N_ * 4);
  float* deginv  = (float*)carve((size_t)N_ * 4);
  int*   csr_src = (int*)carve((size_t)E_ * 4);
  int*   csr_eid = (int*)carve((size_t)E_ * 4);
  (void)ws_size; (void)n_in; (void)out_size;

  const int eb   = (E_ + 255) / 256;
  const int gb4  = (N_ + 3) / 4;   // agg<128>: 4 nodes/block
  const int gb2  = (N_ + 1) / 2;   // agg<256>: 2 nodes/block

  // Reverse-CSR build (edges grouped by destination)
  hipMemsetAsync(cnt, 0, (size_t)N_ * sizeof(int), stream);
  gnet_count<<<eb, 256, 0, stream>>>(colp, cnt, E_);
  gnet_scan<<<1, 256, 0, stream>>>(cnt, rowptr, cursor, deginv, N_);
  gnet_fill<<<eb, 256, 0, stream>>>(rowp, colp, cursor, csr_src, csr_eid, E_);

  // net1 conv1: xn1 = node_attr @ wn1 ; x = relu(seg_sum(xn1[src] + e@we1) + b1)
  launch_gemm<16, 128>(node_attr, wn1, nullptr, nullptr, nullptr, nullptr, W0, N_, 0, stream);
  gnet_agg<128><<<gb4, 128, 0, stream>>>(W0, rowptr, csr_src, csr_eid, edge_attr, we1,
                                         nullptr, b1, nullptr, W1, N_);
  // net1 conv2: y = x @ wn2 ; x_temp = seg_sum(y[src]) + b2 ; x = relu(x_temp)
  launch_gemm<128, 128>(W1, wn2, nullptr, nullptr, nullptr, nullptr, W0, N_, 0, stream);
  gnet_agg<128><<<gb4, 128, 0, stream>>>(W0, rowptr, csr_src, nullptr, nullptr, nullptr,
                                         nullptr, b2, W2, W1, N_);
  // SAGE layer 0 (H -> 2H): m0 = mean(x) ; x1 = relu(m0@sl0 + x@sr0 + sb0)
  gnet_agg<128><<<gb4, 128, 0, stream>>>(W1, rowptr, csr_src, nullptr, nullptr, nullptr,
                                         deginv, nullptr, W0, nullptr, N_);
  launch_gemm<128, 256>(W0, sl0, W1, sr0, sb0, nullptr, W3, N_, 1, stream);
  // SAGE layer 1 (2H -> 2H): m2 = mean(x1) ; x2 = relu(m2@sl1 + x1@sr1 + sb1)
  gnet_agg<256><<<gb2, 128, 0, stream>>>(W3, rowptr, csr_src, nullptr, nullptr, nullptr,
                                         deginv, nullptr, W4, nullptr, N_);
  launch_gemm<256, 256>(W4, sl1, W3, sr1, sb1, nullptr, W4, N_, 1, stream);  // in-place OK (LDS-staged)
  // SAGE layer 2 (2H -> H) + residual: x = relu(m3@sl2 + x2@sr2 + sb2) + x_temp
  gnet_agg<256><<<gb2, 128, 0, stream>>>(W4, rowptr, csr_src, nullptr, nullptr, nullptr,
                                         deginv, nullptr, W3, nullptr, N_);
  launch_gemm<256, 128>(W3, sl2, W4, sr2, sb2, W2, W0, N_, 1, stream);
  // head: h = relu(x@w3a + b3a) ; out = sigmoid(h@w3b + b3b)
  launch_gemm<128, 128>(W0, w3a, nullptr, nullptr, b3a, nullptr, W1, N_, 1, stream);
  launch_gemm<128, 40>(W1, w3b, nullptr, nullptr, b3b, nullptr, (float*)d_out, N_, 2, stream);
}